// GNN_35141422416245
// MI455X (gfx1250) — compile-verified
//
#include <hip/hip_runtime.h>

typedef __attribute__((ext_vector_type(2))) float v2f;
typedef __attribute__((ext_vector_type(8))) float v8f;

constexpr int N = 262144, D = 128, HID = 512, G = 256, M = 4096, OUT = 64;
constexpr int E = 4194304;

// ---- workspace layout (floats) ----
constexpr int OFF_POOLED = 0;
constexpr int OFF_SIZES  = OFF_POOLED + N * D;      // 33554432
constexpr int OFF_COLSUM = OFF_SIZES + G;
constexpr int OFF_GRAM   = OFF_COLSUM + D;
constexpr int OFF_STAT2  = OFF_GRAM + D * D;        // [128 sum | 128 sumsq]
constexpr int OFF_PSUM   = OFF_STAT2 + 2 * D;
constexpr int ZFLOATS    = OFF_PSUM + G * D;        // zero-initialized per call
constexpr int OFF_P      = ZFLOATS;
constexpr int OFF_TMP    = OFF_P + D * D;
constexpr int OFF_SC1    = OFF_TMP + G * D;
constexpr int OFF_SH1    = OFF_SC1 + HID;
constexpr int OFF_SC2    = OFF_SH1 + HID;
constexpr int OFF_SH2    = OFF_SC2 + D;

static __device__ inline v8f wmma4(v2f a, v2f b, v8f c) {
  // D = A(16x4,f32) * B(4x16,f32) + C(16x16,f32)
  return __builtin_amdgcn_wmma_f32_16x16x4_f32(false, a, false, b, (short)0, c,
                                               false, false);
}

// native global f32 atomic add (coarse-grained device memory -> hw fadd)
static __device__ inline void gadd(float* p, float v) { unsafeAtomicAdd(p, v); }

// swizzled z1 index: conflict-free LDS for both writes and GEMM2 v2f reads
static __device__ inline int zidx(int row, int col) {
  int pg = (col >> 1) ^ ((row & 15) << 1);
  return row * 512 + (pg << 1) + (col & 1);
}

// ---------------- zero workspace ----------------
__global__ void zero_kernel(float4* __restrict__ p, int n4) {
  int i = blockIdx.x * blockDim.x + threadIdx.x;
  int stride = gridDim.x * blockDim.x;
  float4 z = {0.f, 0.f, 0.f, 0.f};
  for (; i < n4; i += stride) p[i] = z;
}

// ---------------- graph sizes ----------------
__global__ void sizes_kernel(const int* __restrict__ gidv, float* __restrict__ sizes) {
  int i = blockIdx.x * 256 + threadIdx.x;
  if (i < N) gadd(sizes + gidv[i], 1.0f);
}

// ---------------- edge scatter-add: pooled[dst] += x[src] ----------------
__global__ void scatter_edges(const float* __restrict__ x, const int* __restrict__ ei,
                              float* __restrict__ pooled) {
  long long gid = (long long)blockIdx.x * blockDim.x + threadIdx.x;
  int e = (int)(gid >> 5);
  int lane = (int)(gid & 31);
  if (e >= E) return;
  int src = ei[e];
  int dst = ei[E + e];
  const float4 v = *(const float4*)(x + (long long)src * D + lane * 4);
  float* p = pooled + (long long)dst * D + lane * 4;
  gadd(p + 0, v.x);
  gadd(p + 1, v.y);
  gadd(p + 2, v.z);
  gadd(p + 3, v.w);
}

// ---------------- P = Cl^T @ H  [128x128], K=4096 ----------------
__global__ void P_gemm(const float* __restrict__ Cl, const float* __restrict__ Hm,
                       float* __restrict__ P) {
  int ti = blockIdx.x >> 3, tj = blockIdx.x & 7;
  int lane = threadIdx.x, r = lane & 15, hi = lane >> 4;
  v8f acc = {};
  for (int m0 = 0; m0 < M; m0 += 4) {
    int mm = m0 + 2 * hi;
    v2f a, b;
    a.x = Cl[mm * D + ti * 16 + r];
    a.y = Cl[(mm + 1) * D + ti * 16 + r];
    b.x = Hm[mm * D + tj * 16 + r];
    b.y = Hm[(mm + 1) * D + tj * 16 + r];
    acc = wmma4(a, b, acc);
  }
  for (int v = 0; v < 8; ++v)
    P[(ti * 16 + v + 8 * hi) * D + tj * 16 + r] = acc[v];
}

// ---------------- tmp = Cl[idx] @ P  [256x128], K=128 ----------------
__global__ void tmp_gemm(const float* __restrict__ Cl, const float* __restrict__ P,
                         const int* __restrict__ idxv, float* __restrict__ tmp) {
  int gt = blockIdx.x >> 3, tj = blockIdx.x & 7;
  int lane = threadIdx.x, r = lane & 15, hi = lane >> 4;
  int rowA = idxv[gt * 16 + r];
  v8f acc = {};
  for (int k0 = 0; k0 < D; k0 += 4) {
    int kk = k0 + 2 * hi;
    v2f a, b;
    a.x = Cl[rowA * D + kk];
    a.y = Cl[rowA * D + kk + 1];
    b.x = P[kk * D + tj * 16 + r];
    b.y = P[(kk + 1) * D + tj * 16 + r];
    acc = wmma4(a, b, acc);
  }
  for (int v = 0; v < 8; ++v)
    tmp[(gt * 16 + v + 8 * hi) * D + tj * 16 + r] = acc[v];
}

// ---- pooled = (1+ws0)*pooled + (1+eps)*x + (1+ws1)*tmp[g]/size[g]; colsum ----
__global__ void finalize_pooled(float* __restrict__ pooled, const float* __restrict__ x,
                                const float* __restrict__ tmp, const float* __restrict__ sizes,
                                const int* __restrict__ gidv, const float* __restrict__ eps,
                                const float* __restrict__ wsv, float* __restrict__ colsum) {
  __shared__ float red[256];
  int d = threadIdx.x & 127;
  int rl = threadIdx.x >> 7;  // 0..1
  long long rowbase = (long long)blockIdx.x * 64;
  float w0 = 1.0f + wsv[0];
  float w1 = 1.0f + wsv[1];
  float ep = 1.0f + eps[0];
  float acc = 0.f;
  for (int it = 0; it < 32; ++it) {
    long long row = rowbase + rl + 2 * it;
    int g = gidv[row];
    float val = w0 * pooled[row * D + d] + ep * x[row * D + d] +
                w1 * tmp[g * D + d] / sizes[g];
    pooled[row * D + d] = val;
    acc += val;
  }
  red[threadIdx.x] = acc;
  __syncthreads();
  if (threadIdx.x < 128) gadd(colsum + d, red[threadIdx.x] + red[threadIdx.x + 128]);
}

// ---------------- gram = pooled^T @ pooled  [128x128], K=N (split) ----------------
__global__ void gram_kernel(const float* __restrict__ pooled, float* __restrict__ gram) {
  int tile = blockIdx.x >> 8;   // 0..63
  int split = blockIdx.x & 255; // 0..255, 1024 rows each
  int ta = tile >> 3, tb = tile & 7;
  int lane = threadIdx.x, r = lane & 15, hi = lane >> 4;
  int n0 = split * 1024;
  v8f acc = {};
  for (int k = 0; k < 1024; k += 4) {
    int n = n0 + k + 2 * hi;
    v2f a, b;
    a.x = pooled[(long long)n * D + ta * 16 + r];
    a.y = pooled[(long long)(n + 1) * D + ta * 16 + r];
    b.x = pooled[(long long)n * D + tb * 16 + r];
    b.y = pooled[(long long)(n + 1) * D + tb * 16 + r];
    acc = wmma4(a, b, acc);
  }
  for (int v = 0; v < 8; ++v)
    gadd(gram + (ta * 16 + v + 8 * hi) * D + tb * 16 + r, acc[v]);
}

// ---------------- BN1 scale/shift from Gram trick ----------------
__global__ void bn1_prep(const float* __restrict__ gram, const float* __restrict__ colsum,
                         const float* __restrict__ W1, const float* __restrict__ b1,
                         const float* __restrict__ g1, const float* __restrict__ be1,
                         float* __restrict__ scale1, float* __restrict__ shift1) {
  __shared__ float s1[128], s2[128];
  int j = blockIdx.x;
  int a = threadIdx.x;
  float wa = W1[a * HID + j];
  float inner = 0.f;
  for (int b = 0; b < 128; ++b) inner += gram[a * D + b] * W1[b * HID + j];
  s1[a] = wa * inner;
  s2[a] = (colsum[a] * (1.0f / N)) * wa;
  __syncthreads();
  for (int s = 64; s > 0; s >>= 1) {
    if (a < s) { s1[a] += s1[a + s]; s2[a] += s2[a + s]; }
    __syncthreads();
  }
  if (a == 0) {
    float wGw = s1[0], mw = s2[0];
    float mean = mw + b1[j];
    float ey2 = wGw * (1.0f / N) + 2.f * b1[j] * mw + b1[j] * b1[j];
    float var = ey2 - mean * mean;
    float rs = rsqrtf(var + 1e-5f);
    float sc = rs * g1[j];
    scale1[j] = sc;
    shift1[j] = be1[j] + (b1[j] - mean) * sc;
  }
}

// ---- fused: z = relu(BN1(pooled@W1+b1)) @ W2 + b2 ; accumulate BN2 stats ----
__global__ __launch_bounds__(256) void fused_mlp(
    const float* __restrict__ pooled, const float* __restrict__ W1,
    const float* __restrict__ W2, const float* __restrict__ b2,
    const float* __restrict__ scale1, const float* __restrict__ shift1,
    float* __restrict__ zbuf, float* __restrict__ stat2) {
  // 64KB LDS, time-shared: phase 1 = A tile 32x132 (padded), phase 2 = z1 tile
  // 32x512 (XOR-swizzled), phase 3 = 256-float stats scratch.
  __shared__ float smem[32 * 512];
  int tid = threadIdx.x;
  int lane = tid & 31, w = tid >> 5;
  int r = lane & 15, hi = lane >> 4;
  long long rowbase = (long long)blockIdx.x * 32;

  // ---- stage pooled tile [32 x 128] -> LDS once (row stride 132) ----
  {
    const float4* src = (const float4*)(pooled + rowbase * D);
#pragma unroll
    for (int it = 0; it < 4; ++it) {
      int f = it * 256 + tid;     // float4 index 0..1023
      float4 v = src[f];
      int row = f >> 5;           // 32 float4 per row
      int col4 = f & 31;
      *(float4*)&smem[row * 132 + col4 * 4] = v;
    }
  }
  __syncthreads();

  // ---- GEMM1: Y1 tile [32x512]; wave -> row block (w&1), col chunk (w>>1)*128
  int rw = w & 1;
  int ch = w >> 1;
  v8f acc[8] = {};
  for (int k0 = 0; k0 < 128; k0 += 4) {
    int kk = k0 + 2 * hi;
    v2f a = *(const v2f*)&smem[(rw * 16 + r) * 132 + kk];
#pragma unroll
    for (int t = 0; t < 8; ++t) {
      int col = ch * 128 + t * 16 + r;
      v2f b;
      b.x = W1[kk * HID + col];
      b.y = W1[(kk + 1) * HID + col];
      acc[t] = wmma4(a, b, acc[t]);
    }
  }
  __syncthreads();  // all waves done reading the A tile

  // ---- BN1 + ReLU in registers -> swizzled z1 tile (overwrites A region) ----
#pragma unroll
  for (int t = 0; t < 8; ++t) {
    int col = ch * 128 + t * 16 + r;
    float sc = scale1[col], sh = shift1[col];
#pragma unroll
    for (int v = 0; v < 8; ++v) {
      int rowl = rw * 16 + v + 8 * hi;
      float z1 = acc[t][v] * sc + sh;
      smem[zidx(rowl, col)] = z1 > 0.f ? z1 : 0.f;
    }
  }
  __syncthreads();

  // ---- GEMM2: z tile [32x128]; wave -> row block (w&1), 2 col tiles ----
  int rw2 = w & 1;
  int cg = w >> 1;
  v8f acc2[2] = {};
  for (int k0 = 0; k0 < 512; k0 += 4) {
    int kk = k0 + 2 * hi;
    int rowl = rw2 * 16 + r;
    v2f a = *(const v2f*)&smem[zidx(rowl, kk)];
#pragma unroll
    for (int u = 0; u < 2; ++u) {
      int col = (cg * 2 + u) * 16 + r;
      v2f b;
      b.x = W2[kk * D + col];
      b.y = W2[(kk + 1) * D + col];
      acc2[u] = wmma4(a, b, acc2[u]);
    }
  }
  float ls[2], lq[2];
#pragma unroll
  for (int u = 0; u < 2; ++u) {
    int col = (cg * 2 + u) * 16 + r;
    float bb = b2[col];
    float s = 0.f, q = 0.f;
#pragma unroll
    for (int v = 0; v < 8; ++v) {
      long long row = rowbase + rw2 * 16 + v + 8 * hi;
      float zv = acc2[u][v] + bb;
      zbuf[row * D + col] = zv;
      s += zv;
      q += zv * zv;
    }
    ls[u] = s;
    lq[u] = q;
  }
  __syncthreads();  // done reading z1 tile from smem
  float* ssum = smem;
  float* ssq = smem + 128;
  if (tid < 128) { ssum[tid] = 0.f; ssq[tid] = 0.f; }
  __syncthreads();
#pragma unroll
  for (int u = 0; u < 2; ++u) {
    int col = (cg * 2 + u) * 16 + r;
    atomicAdd(ssum + col, ls[u]);
    atomicAdd(ssq + col, lq[u]);
  }
  __syncthreads();
  if (tid < 128) gadd(stat2 + tid, ssum[tid]);
  else if (tid < 256) gadd(stat2 + tid, ssq[tid - 128]);
}

// ---------------- BN2 scale/shift ----------------
__global__ void bn2_prep(const float* __restrict__ stat2, const float* __restrict__ ge,
                         const float* __restrict__ bee, float* __restrict__ scale2,
                         float* __restrict__ shift2) {
  int d = threadIdx.x;
  float mu = stat2[d] * (1.0f / N);
  float var = stat2[128 + d] * (1.0f / N) - mu * mu;
  float rs = rsqrtf(var + 1e-5f);
  float sc = rs * ge[d];
  scale2[d] = sc;
  shift2[d] = bee[d] - mu * sc;
}

// ---- h = relu(BN2(z)) in place; psum[g] += h (block-reduced) ----
__global__ void final_h(float* __restrict__ hbuf, const float* __restrict__ scale2,
                        const float* __restrict__ shift2, const int* __restrict__ gidv,
                        float* __restrict__ psum) {
  int d = threadIdx.x;
  long long rowbase = (long long)blockIdx.x * 64;
  float sc = scale2[d], sh = shift2[d];
  float acc = 0.f;
  int pg = gidv[rowbase];
  for (int it = 0; it < 64; ++it) {
    long long row = rowbase + it;
    int g = gidv[row];
    if (g != pg) {
      gadd(psum + pg * D + d, acc);
      acc = 0.f;
      pg = g;
    }
    float hv = hbuf[row * D + d] * sc + sh;
    hv = hv > 0.f ? hv : 0.f;
    hbuf[row * D + d] = hv;
    acc += hv;
  }
  gadd(psum + pg * D + d, acc);
}

// ---------------- pooled_h = psum / sizes ----------------
__global__ void pooledh_kernel(const float* __restrict__ psum, const float* __restrict__ sizes,
                               float* __restrict__ ph) {
  int i = blockIdx.x * 256 + threadIdx.x;
  if (i < G * D) ph[i] = psum[i] / sizes[i >> 7];
}

// ---------------- score = pooled_h @ Wp + bp  [256x64], K=128 ----------------
__global__ void score_gemm(const float* __restrict__ ph, const float* __restrict__ Wp,
                           const float* __restrict__ bp, float* __restrict__ score) {
  int gt = blockIdx.x >> 2, jt = blockIdx.x & 3;
  int lane = threadIdx.x, r = lane & 15, hi = lane >> 4;
  v8f acc = {};
  for (int k0 = 0; k0 < D; k0 += 4) {
    int kk = k0 + 2 * hi;
    v2f a, b;
    a.x = ph[(gt * 16 + r) * D + kk];
    a.y = ph[(gt * 16 + r) * D + kk + 1];
    b.x = Wp[kk * OUT + jt * 16 + r];
    b.y = Wp[(kk + 1) * OUT + jt * 16 + r];
    acc = wmma4(a, b, acc);
  }
  int col = jt * 16 + r;
  float bb = bp[col];
  for (int v = 0; v < 8; ++v)
    score[(gt * 16 + v + 8 * hi) * OUT + col] = acc[v] + bb;
}

extern "C" void kernel_launch(void* const* d_in, const int* in_sizes, int n_in,
                              void* d_out, int out_size, void* d_ws, size_t ws_size,
                              hipStream_t stream) {
  (void)in_sizes; (void)n_in; (void)out_size; (void)ws_size;
  const float* x   = (const float*)d_in[0];
  const float* Cl  = (const float*)d_in[1];
  const float* Hm  = (const float*)d_in[2];
  const float* W1  = (const float*)d_in[3];
  const float* b1  = (const float*)d_in[4];
  const float* g1  = (const float*)d_in[5];
  const float* be1 = (const float*)d_in[6];
  const float* W2  = (const float*)d_in[7];
  const float* b2  = (const float*)d_in[8];
  const float* ge  = (const float*)d_in[9];
  const float* bee = (const float*)d_in[10];
  const float* Wp  = (const float*)d_in[11];
  const float* bp  = (const float*)d_in[12];
  const float* eps = (const float*)d_in[13];
  const float* wsv = (const float*)d_in[14];
  const int* ei    = (const int*)d_in[15];
  const int* idxv  = (const int*)d_in[16];
  const int* gidv  = (const int*)d_in[17];

  float* out      = (float*)d_out;
  float* score    = out;
  float* pooled_h = out + G * OUT;
  float* hbuf     = out + G * OUT + G * D;  // z, then h in place

  float* w      = (float*)d_ws;
  float* pooled = w + OFF_POOLED;
  float* sizes  = w + OFF_SIZES;
  float* colsum = w + OFF_COLSUM;
  float* gram   = w + OFF_GRAM;
  float* stat2  = w + OFF_STAT2;
  float* psum   = w + OFF_PSUM;
  float* Pm     = w + OFF_P;
  float* tmpb   = w + OFF_TMP;
  float* sc1    = w + OFF_SC1;
  float* sh1    = w + OFF_SH1;
  float* sc2    = w + OFF_SC2;
  float* sh2    = w + OFF_SH2;

  zero_kernel<<<2048, 256, 0, stream>>>((float4*)w, ZFLOATS / 4);
  sizes_kernel<<<N / 256, 256, 0, stream>>>(gidv, sizes);
  scatter_edges<<<(E * 32) / 256, 256, 0, stream>>>(x, ei, pooled);
  P_gemm<<<64, 32, 0, stream>>>(Cl, Hm, Pm);
  tmp_gemm<<<128, 32, 0, stream>>>(Cl, Pm, idxv, tmpb);
  finalize_pooled<<<N / 64, 256, 0, stream>>>(pooled, x, tmpb, sizes, gidv, eps, wsv, colsum);
  gram_kernel<<<64 * 256, 32, 0, stream>>>(pooled, gram);
  bn1_prep<<<HID, 128, 0, stream>>>(gram, colsum, W1, b1, g1, be1, sc1, sh1);
  fused_mlp<<<N / 32, 256, 0, stream>>>(pooled, W1, W2, b2, sc1, sh1, hbuf, stat2);
  bn2_prep<<<1, 128, 0, stream>>>(stat2, ge, bee, sc2, sh2);
  final_h<<<N / 64, 128, 0, stream>>>(hbuf, sc2, sh2, gidv, psum);
  pooledh_kernel<<<(G * D) / 256, 256, 0, stream>>>(psum, sizes, pooled_h);
  score_gemm<<<64, 32, 0, stream>>>(pooled_h, Wp, bp, score);
}